// ICPLosses_25031069401571
// MI455X (gfx1250) — compile-verified
//
#include <hip/hip_runtime.h>
#include <hip/hip_bf16.h>

// ---------------------------------------------------------------------------
// ICP losses on gfx1250 (MI455X).
//   Phase 1: payload build  : per target j -> {-2tx,-2ty,-2tz, ||t||^2}
//   Phase 2: NN search      : V_WMMA_F32_16X16X4_F32 computes 16x16 score
//                             tiles; running argmin per lane; shfl_xor
//                             butterfly finishes argmin across the 16 lanes
//                             that hold one row.
//   Phase 3: finalize       : gather t/tn at idx, pointwise dv -> out,
//                             block-reduced partial sums for both losses.
//   Phase 4: final reduce   : sum partials, write out[0], out[1].
// ---------------------------------------------------------------------------

typedef __attribute__((ext_vector_type(2))) float v2f;
typedef __attribute__((ext_vector_type(8))) float v8f;

// ---------------- Phase 1: target payload ----------------------------------
__global__ void __launch_bounds__(256)
icp_build_payload(const float* __restrict__ tp, float4* __restrict__ pay, int M) {
    int j = blockIdx.x * blockDim.x + threadIdx.x;
    if (j < M) {
        float x = tp[j], y = tp[M + j], z = tp[2 * M + j];
        pay[j] = make_float4(-2.0f * x, -2.0f * y, -2.0f * z, x * x + y * y + z * z);
    }
}

// ---------------- Phase 2: WMMA nearest-neighbor search --------------------
// One wave owns 16 source rows. A (16x4 f32): lanes 0-15 hold {K0,K1} =
// {sx,sy}; lanes 16-31 hold {K2,K3} = {sz,1}. B (4x16 f32): lanes 0-15 hold
// {-2tx,-2ty} of column j, lanes 16-31 hold {-2tz,||t||^2}. D layout: VGPR r,
// lanes 0-15 -> (row r, col=lane); lanes 16-31 -> (row r+8, col=lane-16).
__global__ void __launch_bounds__(256)
icp_nn_wmma(const float* __restrict__ sp, const float* __restrict__ payf,
            int* __restrict__ idx, int N, int M) {
    const int lane = threadIdx.x & 31;
    const int wave = blockIdx.x * (blockDim.x >> 5) + (threadIdx.x >> 5);
    const int base_i = wave * 16;
    const int hh  = lane >> 4;     // 0: K=0,1 half   1: K=2,3 half
    const int l16 = lane & 15;

    // A fragment (constant for the whole j sweep)
    v2f a;
    {
        const int i = base_i + l16;
        if (hh == 0) { a[0] = sp[i];         a[1] = sp[N + i]; }
        else         { a[0] = sp[2 * N + i]; a[1] = 1.0f;      }
    }

    float best[8];
    int   bestj[8];
#pragma unroll
    for (int r = 0; r < 8; ++r) { best[r] = 3.402823e38f; bestj[r] = 0; }

    // per-lane float offset into the float4 payload stream
    const int laneoff = l16 * 4 + hh * 2;
    const int ntiles  = M >> 4;

    for (int tile = 0; tile < ntiles; ++tile) {
        const float* p = payf + tile * 64 + laneoff;
        __builtin_prefetch((const void*)(p + 8 * 64), 0, 0);  // global_prefetch_b8
        v2f b;
        b[0] = p[0];
        b[1] = p[1];
        v8f c = {};
        c = __builtin_amdgcn_wmma_f32_16x16x4_f32(
                /*neg_a=*/false, a, /*neg_b=*/false, b,
                /*c_mod=*/(short)0, c, /*reuse_a=*/false, /*reuse_b=*/false);
        const int j = tile * 16 + l16;
#pragma unroll
        for (int r = 0; r < 8; ++r) {
            const float v  = c[r];
            const bool  lt = v < best[r];
            best[r]  = lt ? v : best[r];
            bestj[r] = lt ? j : bestj[r];
        }
    }

    // Butterfly argmin across the 16 lanes of each half (xor masks 1,2,4,8
    // stay inside a half). Tie-break on smaller j to match jnp.argmin.
#pragma unroll
    for (int r = 0; r < 8; ++r) {
        float v = best[r];
        int   j = bestj[r];
#pragma unroll
        for (int m = 1; m < 16; m <<= 1) {
            const float ov = __shfl_xor(v, m, 32);
            const int   oj = __shfl_xor(j, m, 32);
            if (ov < v || (ov == v && oj < j)) { v = ov; j = oj; }
        }
        if (l16 == 0) idx[base_i + hh * 8 + r] = j;
    }
}

// ---------------- Phase 3: gather + pointwise + partial loss sums ----------
__global__ void __launch_bounds__(256)
icp_finalize(const float* __restrict__ sp, const float* __restrict__ snrm,
             const float* __restrict__ tp, const float* __restrict__ tnrm,
             const int* __restrict__ idx, float* __restrict__ out,
             float* __restrict__ partials, int N, int M) {
    __shared__ float s0[256];
    __shared__ float s1[256];
    const int i = blockIdx.x * 256 + threadIdx.x;

    const float sx = sp[i], sy = sp[N + i], sz = sp[2 * N + i];
    const int   j  = idx[i];
    const float tx = tp[j], ty = tp[M + j], tz = tp[2 * M + j];
    const float nx = tnrm[j], ny = tnrm[M + j], nz = tnrm[2 * M + j];

    const float dx = sx - tx, dy = sy - ty, dz = sz - tz;
    out[2 + i]         = dx;   // pointwise [1,3,N]: row-major c*N + i
    out[2 + N + i]     = dy;
    out[2 + 2 * N + i] = dz;

    const float d = dx * nx + dy * ny + dz * nz;
    const float p = 1.0f - (snrm[i] * nx + snrm[N + i] * ny + snrm[2 * N + i] * nz);
    s0[threadIdx.x] = d * d;
    s1[threadIdx.x] = p * p;
    __syncthreads();
#pragma unroll
    for (int s = 128; s > 0; s >>= 1) {
        if (threadIdx.x < s) {
            s0[threadIdx.x] += s0[threadIdx.x + s];
            s1[threadIdx.x] += s1[threadIdx.x + s];
        }
        __syncthreads();
    }
    if (threadIdx.x == 0) {
        partials[blockIdx.x]      = s0[0];
        partials[64 + blockIdx.x] = s1[0];
    }
}

// ---------------- Phase 4: final reduction ---------------------------------
__global__ void __launch_bounds__(64)
icp_reduce_final(const float* __restrict__ partials, float* __restrict__ out,
                 int N, int nb) {
    __shared__ float s0[64];
    __shared__ float s1[64];
    const int t = threadIdx.x;
    s0[t] = (t < nb) ? partials[t]      : 0.0f;
    s1[t] = (t < nb) ? partials[64 + t] : 0.0f;
    __syncthreads();
#pragma unroll
    for (int s = 32; s > 0; s >>= 1) {
        if (t < s) { s0[t] += s0[t + s]; s1[t] += s1[t + s]; }
        __syncthreads();
    }
    if (t == 0) {
        out[0] = s0[0] / (float)N;   // loss_po2pl
        out[1] = s1[0] / (float)N;   // loss_pl2pl
    }
}

// ---------------------------------------------------------------------------
extern "C" void kernel_launch(void* const* d_in, const int* in_sizes, int n_in,
                              void* d_out, int out_size, void* d_ws, size_t ws_size,
                              hipStream_t stream) {
    const float* sp   = (const float*)d_in[0];  // source_points   [1,3,N]
    const float* snrm = (const float*)d_in[1];  // source_normals  [1,3,N]
    const float* tp   = (const float*)d_in[2];  // target_points   [1,3,M]
    const float* tnrm = (const float*)d_in[3];  // target_normals  [1,3,M]

    const int N = in_sizes[0] / 3;
    const int M = in_sizes[2] / 3;

    char*   ws       = (char*)d_ws;
    float4* pay      = (float4*)ws;                                   // 16*M bytes
    int*    idx      = (int*)(ws + (size_t)16 * M);                   // 4*N bytes
    float*  partials = (float*)(ws + (size_t)16 * M + (size_t)4 * N); // 128 floats
    float*  out      = (float*)d_out;

    icp_build_payload<<<(M + 255) / 256, 256, 0, stream>>>(tp, pay, M);

    // 1 wave = 16 source rows; 8 waves/block -> N/128 blocks
    icp_nn_wmma<<<N / 128, 256, 0, stream>>>(sp, (const float*)pay, idx, N, M);

    const int nb = N / 256;  // 64 for N=16384
    icp_finalize<<<nb, 256, 0, stream>>>(sp, snrm, tp, tnrm, idx, out, partials, N, M);
    icp_reduce_final<<<1, 64, 0, stream>>>(partials, out, N, nb);
}